// La3_31971736552148
// MI455X (gfx1250) — compile-verified
//
#include <hip/hip_runtime.h>

typedef __attribute__((ext_vector_type(2))) float v2f;
typedef __attribute__((ext_vector_type(8))) float v8f;

#define D_DIM   2000
#define D_VEC   500            // D_DIM / 4 (float4 shift)
#define BLOCK   256            // 8 waves (wave32)
#define GRID_BLOCKS 4096

// Pass 1: grid-stride streaming dot of a[j]*a[j-D] with float4 (b128) loads.
// Wave-level reduction is done with V_WMMA_F32_16X16X4_F32 (see analysis).
__global__ __launch_bounds__(BLOCK) void dot_shift_partial(
    const float* __restrict__ a, float* __restrict__ blockSums, int totalV4)
{
    const float4* __restrict__ A4 = reinterpret_cast<const float4*>(a);
    int v = D_VEC + blockIdx.x * BLOCK + threadIdx.x;
    const int stride = gridDim.x * BLOCK;

    float4 acc = make_float4(0.f, 0.f, 0.f, 0.f);
    #pragma unroll 2
    for (; v < totalV4; v += stride) {
        float4 x = A4[v];            // coalesced 512B per wave
        float4 y = A4[v - D_VEC];    // 8 KB behind -> L2 hit
        acc.x = fmaf(x.x, y.x, acc.x);
        acc.y = fmaf(x.y, y.y, acc.y);
        acc.z = fmaf(x.z, y.z, acc.z);
        acc.w = fmaf(x.w, y.w, acc.w);
    }
    float p = (acc.x + acc.y) + (acc.z + acc.w);

    // ---- wave32 all-lane reduction via V_WMMA_F32_16X16X4_F32 ----
    // A layout (16x4 f32): lanes 0-15 vgpr0 = A[m][0], lanes 16-31 vgpr0 = A[m][2].
    // C1 = P x Ones  =>  C1[m][n] = p[m] + p[m+16]  (q[m], same for every n).
    v2f Ap;   Ap[0] = p;    Ap[1] = 0.0f;
    v2f ones; ones[0] = 1.0f; ones[1] = 1.0f;
    v8f z = {};
    v8f c1 = __builtin_amdgcn_wmma_f32_16x16x4_f32(
        false, Ap, false, ones, (short)0, z, false, false);

    // Feed C1's 8 result VGPRs back as B (4x16) in pairs; A = Ones sums all
    // K rows, accumulating into d. After 4 WMMAs every element of d equals
    // the full 32-lane wave sum.
    v8f d = {};
    v2f b;
    b[0] = c1[0]; b[1] = c1[1];
    d = __builtin_amdgcn_wmma_f32_16x16x4_f32(false, ones, false, b, (short)0, d, false, false);
    b[0] = c1[2]; b[1] = c1[3];
    d = __builtin_amdgcn_wmma_f32_16x16x4_f32(false, ones, false, b, (short)0, d, false, false);
    b[0] = c1[4]; b[1] = c1[5];
    d = __builtin_amdgcn_wmma_f32_16x16x4_f32(false, ones, false, b, (short)0, d, false, false);
    b[0] = c1[6]; b[1] = c1[7];
    d = __builtin_amdgcn_wmma_f32_16x16x4_f32(false, ones, false, b, (short)0, d, false, false);
    float waveSum = d[0];            // identical on all lanes

    __shared__ float wsum[BLOCK / 32];
    const int lane = threadIdx.x & 31;
    const int wid  = threadIdx.x >> 5;
    if (lane == 0) wsum[wid] = waveSum;
    __syncthreads();
    if (threadIdx.x == 0) {
        float s = 0.f;
        #pragma unroll
        for (int i = 0; i < BLOCK / 32; ++i) s += wsum[i];
        blockSums[blockIdx.x] = s;   // one deterministic slot per block
    }
}

// Pass 2: single block, fixed-order deterministic sum + scaling.
__global__ __launch_bounds__(BLOCK) void dot_shift_final(
    const float* __restrict__ blockSums, int nBlocks,
    const float* __restrict__ weight, float* __restrict__ out, float invNm1)
{
    __shared__ float sh[BLOCK];
    float s = 0.f;
    for (int i = threadIdx.x; i < nBlocks; i += BLOCK) s += blockSums[i];
    sh[threadIdx.x] = s;
    __syncthreads();
    if (threadIdx.x == 0) {
        float t = 0.f;
        for (int i = 0; i < BLOCK; ++i) t += sh[i];
        out[0] = -weight[0] * t * invNm1;
    }
}

extern "C" void kernel_launch(void* const* d_in, const int* in_sizes, int n_in,
                              void* d_out, int out_size, void* d_ws, size_t ws_size,
                              hipStream_t stream) {
    const float* a = (const float*)d_in[0];   // factor, N*D fp32
    const float* w = (const float*)d_in[1];   // weight scalar
    float* out = (float*)d_out;

    const long long total = (long long)in_sizes[0];   // N*D = 131,072,000
    const int n = (int)(total / D_DIM);               // N = 65536
    const int totalV4 = (int)(total / 4);

    float* blockSums = (float*)d_ws;                  // GRID_BLOCKS floats (16 KB)

    dot_shift_partial<<<GRID_BLOCKS, BLOCK, 0, stream>>>(a, blockSums, totalV4);
    dot_shift_final<<<1, BLOCK, 0, stream>>>(blockSums, GRID_BLOCKS, w, out,
                                             1.0f / (float)(n - 1));
}